// PCENTransform_83107617178259
// MI455X (gfx1250) — compile-verified
//
#include <hip/hip_runtime.h>

typedef __attribute__((ext_vector_type(2))) float    v2f;
typedef __attribute__((ext_vector_type(8))) float    v8f;
typedef __attribute__((ext_vector_type(4))) unsigned u32x4;
typedef __attribute__((ext_vector_type(8))) int      i32x8;
typedef __attribute__((ext_vector_type(4))) int      i32x4;

#define T_LEN      4000
#define F_LEN      128
#define TILES      (T_LEN / 16)   // 250
#define NCHUNK     4
#define WARM_TILES 40             // a^640 ~ 9e-8 -> carry reconstruction below f32 noise
#define LDS_STRIDE 140            // 128 data + 12 pad dwords (bank-conflict free for both read patterns)
#define S_COEF     0.025f
#define A_COEF     0.975f
#define L2A        (-0.0365258958f)   // log2(0.975)

__global__ __launch_bounds__(256) void pcen_wmma_tdm_kernel(const float* __restrict__ x,
                                                            float* __restrict__ out,
                                                            int Bn)
{
    __shared__ float sm[2 * 16 * LDS_STRIDE];   // double-buffered 16x(128+12) f32 tiles

    const int  tid  = threadIdx.x;
    const int  lane = tid & 31;
    const int  wave = tid >> 5;
    const int  m16  = lane & 15;
    const bool hi   = lane >= 16;
    const int  b    = blockIdx.x;
    const int  c    = blockIdx.y;

    // ---- T-chunking with warm-up (parallel scan via EMA decay) ----
    const int start_tile = (c * TILES) / NCHUNK;
    const int end_tile   = ((c + 1) * TILES) / NCHUNK;
    const int warm       = (c == 0) ? 0 : WARM_TILES;
    const int first_tile = start_tile - warm;
    const int nt         = end_tile - first_tile;

    // a^e without dynamically-indexed arrays (stay in registers)
    auto apow = [](int e) { return __builtin_amdgcn_exp2f(L2A * (float)e); };

    // ---- A-matrix: s*L in four 16x4 chunks (L[j,i] = a^(j-i), lower-tri Toeplitz) ----
    // 32-bit A 16x4 layout: lanes hold row m16; VGPR0 = col(+0 lo / +2 hi), VGPR1 = col(+1 lo / +3 hi)
    v2f Amat[4];
#pragma unroll
    for (int k = 0; k < 4; ++k) {
        int c0 = 4 * k + (hi ? 2 : 0);
        int c1 = c0 + 1;
        Amat[k].x = (m16 >= c0) ? S_COEF * apow(m16 - c0) : 0.0f;
        Amat[k].y = (m16 >= c1) ? S_COEF * apow(m16 - c1) : 0.0f;
    }
    // carry coefficients a^(j+1): C/D row j = r (lanes 0-15) or r+8 (lanes 16-31)
    float coef[8];
#pragma unroll
    for (int r = 0; r < 8; ++r) coef[r] = apow((hi ? r + 9 : r + 1));

    const int       fg    = wave * 16 + m16;          // this lane's frequency channel
    const long long xrow0 = (long long)b * T_LEN;     // first row of this batch
    const int       total_rows = Bn * T_LEN;

    // initial carry: C = x[t0] makes M[t0] = a*C + s*x[t0] = x[t0]  (exact for chunk 0)
    float carry = x[(xrow0 + (long long)first_tile * 16) * F_LEN + fg];

    // ---- TDM: one tensor_load_to_lds per 16x128 tile, with LDS padding 12dw/128dw ----
    auto issue_tdm = [&](int tile_i, int buf) {
        int row0 = (int)xrow0 + (first_tile + tile_i) * 16;
        unsigned long long gaddr =
            (unsigned long long)(const void*)(x + (long long)row0 * F_LEN);
        // generic LDS pointer: low 32 bits are the LDS byte offset (ISA aperture rule)
        unsigned ldsb =
            (unsigned)(unsigned long long)(void*)(&sm[buf * 16 * LDS_STRIDE]);
        unsigned dim1 = (unsigned)(total_rows - row0);

        u32x4 g0;
        g0[0] = 1u;                                       // count=1 (valid), user mode
        g0[1] = ldsb;                                     // lds_addr
        g0[2] = (unsigned)(gaddr & 0xFFFFFFFFull);        // global_addr[31:0]
        g0[3] = (unsigned)((gaddr >> 32) & 0x1FFFFFFu) | (2u << 30); // addr[56:32] | type=2

        i32x8 g1;
        g1[0] = (int)((2u << 16) | (1u << 20) | (6u << 22) | (11u << 25));
                 // data_size=4B | pad_enable | pad_interval=128dw | pad_amount=12dw
        g1[1] = (int)(128u << 16);                        // tensor_dim0 = 128 (bits 79:48 lo)
        g1[2] = (int)((dim1 & 0xFFFFu) << 16);            // dim0 hi=0 | tensor_dim1 lo16
        g1[3] = (int)((dim1 >> 16) | (128u << 16));       // tensor_dim1 hi16 | tile_dim0=128
        g1[4] = 16;                                       // tile_dim1 = 16 rows
        g1[5] = 128;                                      // tensor_dim0_stride = 128 elems
        g1[6] = 0;
        g1[7] = 0;
        i32x4 z4 = {0, 0, 0, 0};
        i32x8 z8 = {0, 0, 0, 0, 0, 0, 0, 0};
        __builtin_amdgcn_tensor_load_to_lds(g0, g1, z4, z4, z8, 0);
    };

    if (wave == 0 && nt > 0) issue_tdm(0, 0);

    for (int i = 0; i < nt; ++i) {
        const int cur = i & 1;
        if (wave == 0) {
            if (i + 1 < nt) {
                issue_tdm(i + 1, (i + 1) & 1);            // prefetch next tile
                __builtin_amdgcn_s_wait_tensorcnt((short)1);
            } else {
                __builtin_amdgcn_s_wait_tensorcnt((short)0);
            }
        }
        __syncthreads();                                  // tile i visible to all waves

        const float* sb = &sm[cur * 16 * LDS_STRIDE];

        // ---- Y = (s*L) @ X via 4x V_WMMA_F32_16X16X4_F32 ----
        v8f acc = {0.f, 0.f, 0.f, 0.f, 0.f, 0.f, 0.f, 0.f};
#pragma unroll
        for (int k = 0; k < 4; ++k) {
            int r0 = 4 * k + (hi ? 2 : 0);                // B 4x16: K even/odd split like A
            v2f Bk;
            Bk.x = sb[(r0)     * LDS_STRIDE + fg];
            Bk.y = sb[(r0 + 1) * LDS_STRIDE + fg];
            acc  = __builtin_amdgcn_wmma_f32_16x16x4_f32(false, Amat[k], false, Bk,
                                                         (short)0, acc, false, false);
        }

        // ---- carry injection: M[j] = Y[j] + a^(j+1)*carry ; next carry = M[15] ----
        float mv[8];
#pragma unroll
        for (int r = 0; r < 8; ++r) mv[r] = acc[r] + coef[r] * carry;
        float nc = __shfl(mv[7], 16 + m16, 32);           // row 15 lives in lanes 16-31

        if (i >= warm) {                                  // skip stores during warm-up
            const int t0 = (first_tile + i) * 16;
#pragma unroll
            for (int r = 0; r < 8; ++r) {
                int   tr = r + (hi ? 8 : 0);
                float xv = sb[tr * LDS_STRIDE + fg];
                float me = mv[r] + 1e-6f;
                // (M+eps)^0.98 via raw v_log_f32 / v_exp_f32 (me is always normal, positive)
                float p  = __builtin_amdgcn_exp2f(0.98f * __builtin_amdgcn_logf(me));
                float v  = xv * __builtin_amdgcn_rcpf(p) + 2.0f;
                out[(xrow0 + t0 + tr) * F_LEN + fg] =
                    __builtin_amdgcn_sqrtf(v) - 1.41421356237f;   // - 2^0.5
            }
        }
        carry = nc;
        __syncthreads();                                  // safe to overwrite buffer at i+2
    }
}

extern "C" void kernel_launch(void* const* d_in, const int* in_sizes, int n_in,
                              void* d_out, int out_size, void* d_ws, size_t ws_size,
                              hipStream_t stream) {
    const float* x   = (const float*)d_in[0];
    float*       out = (float*)d_out;
    const int    Bn  = in_sizes[0] / (T_LEN * F_LEN);
    dim3 grid(Bn, NCHUNK);
    pcen_wmma_tdm_kernel<<<grid, 256, 0, stream>>>(x, out, Bn);
}